// Net_29025388986627
// MI455X (gfx1250) — compile-verified
//
#include <hip/hip_runtime.h>

#define N_NODES 100000
#define N_EDGES 1600000
#define IN_FEATS 16
#define H_FEAT 128
#define E_FEATS 20
#define NUM_CLASSES 4
#define NUM_LAYERS 20
#define NEG_INF -1000000000.0f
#define SLOPE 0.01f
#define SCAN_BLK 1024

typedef __attribute__((ext_vector_type(2))) float v2f;
typedef __attribute__((ext_vector_type(8))) float v8f;

__device__ __forceinline__ float leakyf(float v) {
    return v >= 0.0f ? v : SLOPE * v;
}

// gfx1250 async global->LDS copy (ASYNCcnt-tracked), 16B per lane
__device__ __forceinline__ void async_copy_b128(unsigned int lds_addr, unsigned long long gaddr) {
    asm volatile("global_load_async_to_lds_b128 %0, %1, off"
                 :: "v"(lds_addr), "v"(gaddr)
                 : "memory");
}

__device__ __forceinline__ void wait_asynccnt0() {
    asm volatile("s_wait_asynccnt 0x0" ::: "memory");
}

// ---------------- CSR construction (once per launch) ----------------

__global__ void zero_counts_kernel(int* __restrict__ counts) {
    int i = blockIdx.x * blockDim.x + threadIdx.x;
    if (i < N_NODES) counts[i] = 0;
}

__global__ void count_edges_kernel(const int* __restrict__ dst, int* __restrict__ counts) {
    int e = blockIdx.x * blockDim.x + threadIdx.x;
    if (e < N_EDGES) atomicAdd(&counts[dst[e]], 1);
}

// per-block exclusive scan (Hillis-Steele in LDS), emits block sums
__global__ void __launch_bounds__(SCAN_BLK)
scan1_kernel(const int* __restrict__ counts, int* __restrict__ row_ptr, int* __restrict__ blksum) {
    __shared__ int s[SCAN_BLK];
    const int tid = threadIdx.x;
    const int i = blockIdx.x * SCAN_BLK + tid;
    const int v = (i < N_NODES) ? counts[i] : 0;
    s[tid] = v;
    __syncthreads();
    for (int off = 1; off < SCAN_BLK; off <<= 1) {
        int t = (tid >= off) ? s[tid - off] : 0;
        __syncthreads();
        s[tid] += t;
        __syncthreads();
    }
    if (i < N_NODES) row_ptr[i] = s[tid] - v;  // exclusive within block
    if (tid == SCAN_BLK - 1) blksum[blockIdx.x] = s[tid];
}

__global__ void scan2_kernel(const int* __restrict__ blksum, int* __restrict__ blkoff, int nb) {
    if (blockIdx.x == 0 && threadIdx.x == 0) {
        int run = 0;
        for (int i = 0; i < nb; ++i) { blkoff[i] = run; run += blksum[i]; }
    }
}

__global__ void scan3_kernel(int* __restrict__ row_ptr, const int* __restrict__ blkoff) {
    int i = blockIdx.x * blockDim.x + threadIdx.x;
    if (i < N_NODES) row_ptr[i] += blkoff[i / SCAN_BLK];
    if (i == 0) row_ptr[N_NODES] = N_EDGES;
}

__global__ void copy_cursor_kernel(const int* __restrict__ row_ptr, int* __restrict__ cursor) {
    int i = blockIdx.x * blockDim.x + threadIdx.x;
    if (i < N_NODES) cursor[i] = row_ptr[i];
}

__global__ void fill_csr_kernel(const int* __restrict__ src, const int* __restrict__ dst,
                                const float* __restrict__ edge_attr, int* __restrict__ cursor,
                                int* __restrict__ csr_src, float* __restrict__ csr_w) {
    int e = blockIdx.x * blockDim.x + threadIdx.x;
    if (e >= N_EDGES) return;
    int d = dst[e];
    int pos = atomicAdd(&cursor[d], 1);
    csr_src[pos] = src[e];
    csr_w[pos] = edge_attr[(long long)e * E_FEATS];  // only feature 0 used
}

// ---------------- pull-mode max aggregation (no atomics) ----------------
// block = 128 threads; (128/K) nodes per block, one thread per (node, feat)
template <int K>
__global__ void __launch_bounds__(128)
agg_kernel(const float* __restrict__ h, const int* __restrict__ row_ptr,
           const int* __restrict__ csr_src, const float* __restrict__ csr_w,
           float* __restrict__ agg) {
    const int NPB = 128 / K;
    const int node = blockIdx.x * NPB + threadIdx.x / K;
    const int f = threadIdx.x % K;
    if (node >= N_NODES) return;
    const int beg = row_ptr[node];
    const int end = row_ptr[node + 1];
    float acc = NEG_INF;
    int i = beg;
    for (; i + 1 < end; i += 2) {  // 2 edges in flight to hide L2 latency
        int s0 = csr_src[i], s1 = csr_src[i + 1];
        float w0 = csr_w[i], w1 = csr_w[i + 1];
        float m0 = h[(long long)s0 * K + f] * w0;
        float m1 = h[(long long)s1 * K + f] * w1;
        acc = fmaxf(acc, fmaxf(m0, m1));
    }
    if (i < end) {
        int s0 = csr_src[i];
        acc = fmaxf(acc, h[(long long)s0 * K + f] * csr_w[i]);
    }
    agg[(long long)node * K + f] = (acc == NEG_INF) ? 0.0f : acc;
}

// ---------------- GIN MLP via fp32 WMMA + async W2 staging ----------------
// Per block: 4 waves, 16 nodes each (64 nodes/block).
// out = leaky( leaky(z @ W1 + b1) @ W2 + b2 ),  z = (1+eps)*h + agg
// Weights are pre-packed pairwise: Wp[(k>>1)*256 + col*2 + (k&1)] = W[k][col]
// Dynamic LDS layout: zs[4*16*K_IN] | ts[4*16*128] | w2[128*128]
template <int K_IN>
__global__ void __launch_bounds__(128)
gin_mlp_kernel(const float* __restrict__ h,     // N x K_IN
               const float* __restrict__ agg,   // N x K_IN
               const float* __restrict__ Wp1,   // packed K_IN x 128
               const float* __restrict__ b1,    // 128
               const float* __restrict__ Wp2,   // packed 128 x 128
               const float* __restrict__ b2,    // 128
               const float* __restrict__ epsp,  // scalar
               float* __restrict__ out) {       // N x 128
    extern __shared__ __align__(16) float smem[];
    float* zs = smem;                         // 4*16*K_IN
    float* ts = zs + 4 * 16 * K_IN;           // 4*16*H_FEAT
    float* w2 = ts + 4 * 16 * H_FEAT;         // H_FEAT*H_FEAT

    const int wave = threadIdx.x >> 5;
    const int lane = threadIdx.x & 31;
    const int half = lane >> 4;  // 0: lanes 0-15, 1: lanes 16-31
    const int m    = lane & 15;

    const long long rowBase = (long long)blockIdx.x * 64 + wave * 16;
    const bool active = rowBase < N_NODES;  // N_NODES % 16 == 0 -> full tiles

    const float eps1 = 1.0f + epsp[0];

    float* z = zs + wave * (16 * K_IN);
    float* t = ts + wave * (16 * H_FEAT);

    // ---- kick off async copy of packed W2 (64KB) into LDS; overlaps with
    // z staging and matmul 1; LDS offset = low 32 bits of flat address ----
    {
        const unsigned int ldsBase = (unsigned int)(unsigned long long)(uintptr_t)(void*)w2;
        const unsigned long long gBase = (unsigned long long)(uintptr_t)(const void*)Wp2;
        const int total = H_FEAT * H_FEAT * 4;  // bytes
        for (int off = threadIdx.x * 16; off < total; off += 128 * 16) {
            async_copy_b128(ldsBase + (unsigned int)off, gBase + (unsigned long long)off);
        }
    }

    // warm L0/L2 for W1 (gfx1250 global_prefetch_b8)
    __builtin_prefetch((const char*)Wp1 + (threadIdx.x << 7), 0, 3);

    if (active) {
        // stage z tile (16 x K_IN) in LDS, float4-wide
        const float4* h4 = (const float4*)h;
        const float4* a4 = (const float4*)agg;
        float4* z4 = (float4*)z;
        const int KV = K_IN / 4;
        for (int idx = lane; idx < 16 * KV; idx += 32) {
            int r = idx / KV;
            int k4 = idx - r * KV;
            long long g = (rowBase + r) * (long long)KV + k4;
            float4 av = a4[g];
            float4 hv = h4[g];
            float4 zv;
            zv.x = eps1 * hv.x + av.x;
            zv.y = eps1 * hv.y + av.y;
            zv.z = eps1 * hv.z + av.z;
            zv.w = eps1 * hv.w + av.w;
            z4[idx] = zv;
        }
    }
    __syncthreads();

    if (active) {
        // ---- matmul 1: t = leaky(z @ W1 + b1), K = K_IN, B from global ----
        const float2* B1 = (const float2*)Wp1;
        v8f acc[8] = {};
        for (int ks = 0; ks < K_IN / 4; ++ks) {
            const int k = ks * 4 + 2 * half;  // A: lanes<16 hold K=k,k+1; lanes>=16 K=k+2,k+3
            const int kp = ks * 2 + half;     // packed pair row
            v2f a;
            a.x = z[m * K_IN + k];
            a.y = z[m * K_IN + k + 1];
#pragma unroll
            for (int nt = 0; nt < 8; ++nt) {
                const int col = nt * 16 + m;
                float2 bb = B1[kp * H_FEAT + col];  // single b64 load
                v2f b; b.x = bb.x; b.y = bb.y;
                acc[nt] = __builtin_amdgcn_wmma_f32_16x16x4_f32(
                    false, a, false, b, (short)0, acc[nt], false, false);
            }
        }
        // C/D layout: VGPR r holds row r (lanes 0-15) / row r+8 (lanes 16-31), col = m
#pragma unroll
        for (int nt = 0; nt < 8; ++nt) {
            const int col = nt * 16 + m;
            const float bias = b1[col];
#pragma unroll
            for (int r = 0; r < 8; ++r) {
                const int row = r + 8 * half;
                t[row * H_FEAT + col] = leakyf(acc[nt][r] + bias);
            }
        }
    }

    // async W2 copy must be complete and visible to all waves
    wait_asynccnt0();
    __syncthreads();

    if (active) {
        // ---- matmul 2: out = leaky(t @ W2 + b2), K = 128, B from LDS ----
        const float2* B2 = (const float2*)w2;
        v8f acc[8] = {};
        for (int ks = 0; ks < H_FEAT / 4; ++ks) {
            const int k = ks * 4 + 2 * half;
            const int kp = ks * 2 + half;
            v2f a;
            a.x = t[m * H_FEAT + k];
            a.y = t[m * H_FEAT + k + 1];
#pragma unroll
            for (int nt = 0; nt < 8; ++nt) {
                const int col = nt * 16 + m;
                float2 bb = B2[kp * H_FEAT + col];  // ds_load_b64
                v2f b; b.x = bb.x; b.y = bb.y;
                acc[nt] = __builtin_amdgcn_wmma_f32_16x16x4_f32(
                    false, a, false, b, (short)0, acc[nt], false, false);
            }
        }
#pragma unroll
        for (int nt = 0; nt < 8; ++nt) {
            const int col = nt * 16 + m;
            const float bias = b2[col];
#pragma unroll
            for (int r = 0; r < 8; ++r) {
                const int row = r + 8 * half;
                out[(rowBase + row) * (long long)H_FEAT + col] = leakyf(acc[nt][r] + bias);
            }
        }
    }
}

// pack W (rows x 128) -> pair-interleaved: Wp[(k>>1)*256 + col*2 + (k&1)]
__global__ void pack_weights_kernel(const float* __restrict__ W, float* __restrict__ Wp, int rows) {
    int idx = blockIdx.x * blockDim.x + threadIdx.x;
    if (idx >= rows * H_FEAT) return;
    int k = idx / H_FEAT;
    int col = idx - k * H_FEAT;
    Wp[(k >> 1) * (2 * H_FEAT) + col * 2 + (k & 1)] = W[idx];
}

// final: out[n][c] = h[n] . fcW[:,c] + fcb[c]
__global__ void classifier_kernel(const float* __restrict__ h,
                                  const float* __restrict__ fcW,  // 128 x 4
                                  const float* __restrict__ fcb,  // 4
                                  float* __restrict__ out) {      // N x 4
    long long tid = (long long)blockIdx.x * blockDim.x + threadIdx.x;
    if (tid >= (long long)N_NODES * NUM_CLASSES) return;
    int n = (int)(tid / NUM_CLASSES);
    int c = (int)(tid - (long long)n * NUM_CLASSES);
    const float* hr = h + (long long)n * H_FEAT;
    float s = fcb[c];
#pragma unroll 8
    for (int k = 0; k < H_FEAT; ++k) s += hr[k] * fcW[k * NUM_CLASSES + c];
    out[tid] = s;
}

extern "C" void kernel_launch(void* const* d_in, const int* in_sizes, int n_in,
                              void* d_out, int out_size, void* d_ws, size_t ws_size,
                              hipStream_t stream) {
    const float* x         = (const float*)d_in[0];
    const float* edge_attr = (const float*)d_in[1];
    const float* W1_0      = (const float*)d_in[2];
    const float* b1_0      = (const float*)d_in[3];
    const float* W2_0      = (const float*)d_in[4];
    const float* b2_0      = (const float*)d_in[5];
    const float* eps0      = (const float*)d_in[6];
    const float* W1s       = (const float*)d_in[7];
    const float* b1s       = (const float*)d_in[8];
    const float* W2s       = (const float*)d_in[9];
    const float* b2s       = (const float*)d_in[10];
    const float* epss      = (const float*)d_in[11];
    const float* fcW       = (const float*)d_in[12];
    const float* fcb       = (const float*)d_in[13];
    const int*   edge_index = (const int*)d_in[14];
    const int* srcI = edge_index;            // edge_index[0]
    const int* dstI = edge_index + N_EDGES;  // edge_index[1]

    // ---- workspace layout ----
    float* hA  = (float*)d_ws;
    float* hB  = hA + (size_t)N_NODES * H_FEAT;
    float* agg = hB + (size_t)N_NODES * H_FEAT;
    int* row_ptr = (int*)(agg + (size_t)N_NODES * H_FEAT);  // N_NODES+1
    int* counts  = row_ptr + (N_NODES + 4);
    int* cursor  = counts + N_NODES;
    int* blksum  = cursor + N_NODES;   // <=128 entries
    int* blkoff  = blksum + 128;       // <=128 entries
    int* csr_src = blkoff + 128;       // N_EDGES
    float* csr_w = (float*)(csr_src + N_EDGES);  // N_EDGES
    float* wpack = csr_w + N_EDGES;    // packed weights: 2048 + 16384 + 19*32768

    float* wp1_0 = wpack;                 // 16x128
    float* wp2_0 = wp1_0 + 16 * H_FEAT;   // 128x128
    float* wp_layers = wp2_0 + H_FEAT * H_FEAT;

    const int nb_scan = (N_NODES + SCAN_BLK - 1) / SCAN_BLK;
    const int mlp_grid = (N_NODES + 63) / 64;

    const size_t shmem16 =
        (size_t)(4 * 16 * IN_FEATS + 4 * 16 * H_FEAT + H_FEAT * H_FEAT) * sizeof(float);
    const size_t shmem128 =
        (size_t)(4 * 16 * H_FEAT + 4 * 16 * H_FEAT + H_FEAT * H_FEAT) * sizeof(float);

    // ---- pack weights (once per launch) ----
    {
        int n0 = IN_FEATS * H_FEAT;
        pack_weights_kernel<<<(n0 + 255) / 256, 256, 0, stream>>>(W1_0, wp1_0, IN_FEATS);
        int n1 = H_FEAT * H_FEAT;
        pack_weights_kernel<<<(n1 + 255) / 256, 256, 0, stream>>>(W2_0, wp2_0, H_FEAT);
        for (int l = 0; l < NUM_LAYERS - 1; ++l) {
            float* wp1 = wp_layers + (size_t)l * 2 * H_FEAT * H_FEAT;
            float* wp2 = wp1 + H_FEAT * H_FEAT;
            pack_weights_kernel<<<(n1 + 255) / 256, 256, 0, stream>>>(
                W1s + (size_t)l * H_FEAT * H_FEAT, wp1, H_FEAT);
            pack_weights_kernel<<<(n1 + 255) / 256, 256, 0, stream>>>(
                W2s + (size_t)l * H_FEAT * H_FEAT, wp2, H_FEAT);
        }
    }

    // ---- build CSR by destination (once per launch) ----
    zero_counts_kernel<<<(N_NODES + 255) / 256, 256, 0, stream>>>(counts);
    count_edges_kernel<<<(N_EDGES + 255) / 256, 256, 0, stream>>>(dstI, counts);
    scan1_kernel<<<nb_scan, SCAN_BLK, 0, stream>>>(counts, row_ptr, blksum);
    scan2_kernel<<<1, 1, 0, stream>>>(blksum, blkoff, nb_scan);
    scan3_kernel<<<(N_NODES + 255) / 256, 256, 0, stream>>>(row_ptr, blkoff);
    copy_cursor_kernel<<<(N_NODES + 255) / 256, 256, 0, stream>>>(row_ptr, cursor);
    fill_csr_kernel<<<(N_EDGES + 255) / 256, 256, 0, stream>>>(
        srcI, dstI, edge_attr, cursor, csr_src, csr_w);

    // ---- layer 0 (K_IN = 16) ----
    agg_kernel<IN_FEATS><<<(N_NODES * IN_FEATS + 127) / 128, 128, 0, stream>>>(
        x, row_ptr, csr_src, csr_w, agg);
    gin_mlp_kernel<IN_FEATS><<<mlp_grid, 128, shmem16, stream>>>(
        x, agg, wp1_0, b1_0, wp2_0, b2_0, eps0, hA);

    // ---- layers 1..19 (K_IN = 128) ----
    float* cur = hA;
    float* nxt = hB;
    for (int l = 0; l < NUM_LAYERS - 1; ++l) {
        agg_kernel<H_FEAT><<<N_NODES, 128, 0, stream>>>(cur, row_ptr, csr_src, csr_w, agg);
        float* wp1 = wp_layers + (size_t)l * 2 * H_FEAT * H_FEAT;
        float* wp2 = wp1 + H_FEAT * H_FEAT;
        gin_mlp_kernel<H_FEAT><<<mlp_grid, 128, shmem128, stream>>>(
            cur, agg, wp1, b1s + (size_t)l * H_FEAT,
            wp2, b2s + (size_t)l * H_FEAT, epss + l, nxt);
        float* tmp = cur; cur = nxt; nxt = tmp;
    }

    // ---- classifier ----
    long long th = (long long)N_NODES * NUM_CLASSES;
    classifier_kernel<<<(int)((th + 255) / 256), 256, 0, stream>>>(
        cur, fcW, fcb, (float*)d_out);
}